// Conv2d_19499151524307
// MI455X (gfx1250) — compile-verified
//
#include <hip/hip_runtime.h>

typedef __attribute__((ext_vector_type(2))) float v2f;
typedef __attribute__((ext_vector_type(4))) float f4;
typedef __attribute__((ext_vector_type(8))) float v8f;

#define TILES_PER_WAVE  4
#define WAVES_PER_BLOCK 8

// out[w] = dot(x[w*49 .. w*49+48], wgt[0..48]) + bias[0]
// 16 windows per tile, 4 tiles per wave32, 13 x V_WMMA_F32_16X16X4_F32 / tile.
// Weight B-fragments are register-resident across the tile loop.
__global__ __launch_bounds__(256) void conv2d_im2col_wmma_f32(
    const float* __restrict__ x,     // [totalW * 49]
    const float* __restrict__ wgt,   // [49]
    const float* __restrict__ bias,  // [1]
    float* __restrict__ out,         // [totalW]
    long long totalW, long long totalTiles)
{
    const int  lane  = threadIdx.x & 31;
    const int  wave  = threadIdx.x >> 5;
    const long long group = (long long)blockIdx.x * WAVES_PER_BLOCK + wave;
    const long long tile0 = group * TILES_PER_WAVE;
    if (tile0 >= totalTiles) return;               // wave-uniform exit

    const bool hiL = lane >= 16;
    const int  m   = lane & 15;                    // A-matrix row fed by lane
    const float bb = bias[0];

    // ---- weight B-fragments (4x16, w replicated over N), loaded ONCE ----
    // low lanes hold K={4s,4s+1}, high lanes K={4s+2,4s+3}
    v2f bw[13];
    #pragma unroll
    for (int s = 0; s < 12; ++s) {
        const int ka = 4 * s + (hiL ? 2 : 0);
        bw[s].x = wgt[ka];
        bw[s].y = wgt[ka + 1];
    }
    // step 12 covers K=48..51; only K=48 exists -> zero the rest via B
    bw[12].x = hiL ? 0.0f : wgt[48];
    bw[12].y = 0.0f;

    const long long tileEnd = (tile0 + TILES_PER_WAVE < totalTiles)
                                  ? (tile0 + TILES_PER_WAVE) : totalTiles;

    for (long long tile = tile0; tile < tileEnd; ++tile) {
        const long long w0 = tile * 16;            // first window of tile

        if (w0 + 16 <= totalW) {
            // ---- full tile: branchless WMMA path (EXEC all-ones) ----
            const float* __restrict__ px = x + (w0 + m) * 49ll;

            if (tile + 1 < tileEnd) {              // overlap next tile's fetch
                const float* pn = px + 16ll * 49ll;
                __builtin_prefetch(pn, 0, 0);      // global_prefetch_b8
                __builtin_prefetch(pn + 32, 0, 0);
            }

            v8f c = {};
            #pragma unroll
            for (int s = 0; s < 12; ++s) {
                const int ka = 4 * s + (hiL ? 2 : 0);
                v2f a;                              // b64-combinable pair
                a.x = px[ka];
                a.y = px[ka + 1];
                c = __builtin_amdgcn_wmma_f32_16x16x4_f32(
                        false, a, false, bw[s], (short)0, c, false, false);
            }
            {   // K=48 remainder: hi-lane A garbage is multiplied by B==0
                v2f a;
                a.x = px[48];                       // always in-bounds
                a.y = 0.0f;
                c = __builtin_amdgcn_wmma_f32_16x16x4_f32(
                        false, a, false, bw[12], (short)0, c, false, false);
            }

            // C layout: lane 0 -> rows 0..7, lane 16 -> rows 8..15
            // (all columns identical since B is replicated over N)
            if ((lane & 15) == 0) {
                const long long o = w0 + (hiL ? 8 : 0);
                f4 lo = {c[0] + bb, c[1] + bb, c[2] + bb, c[3] + bb};
                f4 hi = {c[4] + bb, c[5] + bb, c[6] + bb, c[7] + bb};
                *(f4*)(out + o)     = lo;          // 16B-aligned b128 stores
                *(f4*)(out + o + 4) = hi;
            }
        } else {
            // ---- ragged tail (not hit for 64*50176) ----
            const long long w = w0 + lane;
            if (lane < 16 && w < totalW) {
                const float* __restrict__ px = x + w * 49ll;
                float acc = 0.0f;
                #pragma unroll
                for (int k = 0; k < 49; ++k) acc = fmaf(px[k], wgt[k], acc);
                out[w] = acc + bb;
            }
        }
    }
}

extern "C" void kernel_launch(void* const* d_in, const int* in_sizes, int n_in,
                              void* d_out, int out_size, void* d_ws, size_t ws_size,
                              hipStream_t stream) {
    const float* x    = (const float*)d_in[0];  // enc_x  [B*WINDOWS*49] f32
    const float* wgt  = (const float*)d_in[1];  // weight [49]           f32
    const float* bias = (const float*)d_in[2];  // bias   [1]            f32
    // d_in[3] = windows_nb scalar (unused; out_size == B*WINDOWS)
    float* out = (float*)d_out;

    const long long totalW     = (long long)out_size;            // 3,211,264
    const long long totalTiles = (totalW + 15) / 16;             // 200,704
    const long long groups     = (totalTiles + TILES_PER_WAVE - 1) / TILES_PER_WAVE;
    const long long nBlocks    = (groups + WAVES_PER_BLOCK - 1) / WAVES_PER_BLOCK;

    conv2d_im2col_wmma_f32<<<dim3((unsigned)nBlocks), dim3(256), 0, stream>>>(
        x, wgt, bias, out, totalW, totalTiles);
}